// NGNN_GCNConv_28398323761561
// MI455X (gfx1250) — compile-verified
//
#include <hip/hip_runtime.h>

// NGNN GCNConv for MI455X (gfx1250): fp32 WMMA GEMMs + run-length edge aggregation.
//
// Roofline: ~16.4 GFLOP of GEMM vs ~300 MB HBM traffic (plus ~1.6 GB of
// L2-resident edge gather/scatter). At 23.3 TB/s the memory floor is ~15-60 us,
// so f32 WMMA (full fp32 accuracy vs the fp32 reference) is the right
// precision: the matrix pipes are nowhere near being the bottleneck.

#define IN_C  256
#define HID_C 128
#define OUT_C 256

typedef int idx_t;  // harness passes integer inputs as int32
typedef __attribute__((ext_vector_type(2))) float v2f;
typedef __attribute__((ext_vector_type(8))) float v8f;

// ---------------------------------------------------------------- utilities
__global__ void zero_f32_kernel(float* __restrict__ p, long n) {
  long i = (long)blockIdx.x * blockDim.x + threadIdx.x;
  const long stride = (long)gridDim.x * blockDim.x;
  for (; i < n; i += stride) p[i] = 0.0f;
}

__global__ void degree_kernel(const idx_t* __restrict__ src,
                              const idx_t* __restrict__ dst,
                              float* __restrict__ deg_out,
                              float* __restrict__ deg_in, int E) {
  const int e = blockIdx.x * blockDim.x + threadIdx.x;
  if (e >= E) return;
  unsafeAtomicAdd(&deg_out[src[e]], 1.0f);
  unsafeAtomicAdd(&deg_in[dst[e]], 1.0f);
}

__global__ void rsqrt_deg_kernel(float* __restrict__ deg, int n) {
  const int i = blockIdx.x * blockDim.x + threadIdx.x;
  if (i >= n) return;
  deg[i] = rsqrtf(fmaxf(deg[i], 1.0f));
}

// W[rows][cols] -> Wt[cols][rows]; puts W_conv into the same [N,K] layout as
// the torch Linear weights so every GEMM B-fragment load is a single b64.
__global__ void transpose_kernel(const float* __restrict__ W,
                                 float* __restrict__ Wt, int rows, int cols) {
  const int i = blockIdx.x * blockDim.x + threadIdx.x;
  if (i >= rows * cols) return;
  const int r = i / cols;
  const int c = i - r * cols;
  Wt[(long)c * rows + r] = W[i];
}

// ------------------------------------------------------- WMMA f32 GEMM wave
// D[m,n] = act( (SCALE_A ? scaleA[m] : 1) * A[m,:] @ B + bias[n] )
// with B[k][n] = W[n*KDIM + k]  ([N,K] layout for all weights).
// One wave computes a 16-row x (NTILES*16)-col strip. Full EXEC always
// (N is a multiple of 16, channel dims multiples of 16*NTILES per grid.y).
template <int KDIM, int NTILES, int NOUT, bool HAS_BIAS, bool RELU,
          bool SCALE_A>
__global__ __launch_bounds__(64) void gemm_wmma_kernel(
    const float* __restrict__ A, const float* __restrict__ W,
    const float* __restrict__ bias, const float* __restrict__ scaleA,
    float* __restrict__ D) {
  const int lane = threadIdx.x & 31;
  const int rowTile = blockIdx.x * 2 + (threadIdx.x >> 5);
  const int m0 = rowTile * 16;
  const int n0 = blockIdx.y * (NTILES * 16);
  const int mrow = m0 + (lane & 15);
  const int khalf = lane >> 4;  // lanes 16..31 carry K+2,K+3 for A and B

  const float s = SCALE_A ? scaleA[mrow] : 1.0f;

  v8f acc[NTILES];
#pragma unroll
  for (int t = 0; t < NTILES; ++t) acc[t] = (v8f){0, 0, 0, 0, 0, 0, 0, 0};

  const float* __restrict__ arow = A + (long)mrow * KDIM;
  for (int k0 = 0; k0 < KDIM; k0 += 4) {
    const int ka = k0 + khalf * 2;  // even -> float2 aligned
    const float2 av = *(const float2*)(arow + ka);
    v2f a;
    a.x = av.x * s;
    a.y = av.y * s;
#pragma unroll
    for (int t = 0; t < NTILES; ++t) {
      const int ncol = n0 + t * 16 + (lane & 15);
      const float2 bv = *(const float2*)(W + (long)ncol * KDIM + ka);
      v2f b;
      b.x = bv.x;
      b.y = bv.y;
      acc[t] = __builtin_amdgcn_wmma_f32_16x16x4_f32(
          false, a, false, b, (short)0, acc[t], false, false);
    }
  }

#pragma unroll
  for (int t = 0; t < NTILES; ++t) {
    const int ncol = n0 + t * 16 + (lane & 15);
#pragma unroll
    for (int v = 0; v < 8; ++v) {
      const int row = m0 + v + 8 * khalf;  // C/D layout: lanes16-31 -> M=v+8
      float val = acc[t][v];
      if constexpr (HAS_BIAS) val += bias[ncol];
      if constexpr (RELU) val = fmaxf(val, 0.0f);
      D[(long)row * NOUT + ncol] = val;
    }
  }
}

// ------------------------------------------------- edge aggregation (sorted dst)
// One wave per chunk of edges; lane owns 4 channels (32*4 = 128).
// dst is sorted, so accumulate runs of equal dst in registers and flush with
// global f32 atomics only on run boundaries (~avg-degree x fewer atomics).
#define EDGES_PER_WAVE 512
__global__ void aggregate_kernel(const float* __restrict__ h1,
                                 const idx_t* __restrict__ src,
                                 const idx_t* __restrict__ dst,
                                 float* __restrict__ m, int E) {
  const int wave = (int)((blockIdx.x * blockDim.x + threadIdx.x) >> 5);
  const int lane = threadIdx.x & 31;
  long e0 = (long)wave * EDGES_PER_WAVE;
  if (e0 >= E) return;
  long e1 = e0 + EDGES_PER_WAVE;
  if (e1 > E) e1 = E;
  const int c = lane * 4;

  float4 acc = make_float4(0.f, 0.f, 0.f, 0.f);
  int cur = dst[e0];  // uniform across wave -> converged control flow
  for (long e = e0; e < e1; ++e) {
    const int d = dst[e];
    if (d != cur) {
      float* p = m + (long)cur * HID_C + c;
      unsafeAtomicAdd(p + 0, acc.x);
      unsafeAtomicAdd(p + 1, acc.y);
      unsafeAtomicAdd(p + 2, acc.z);
      unsafeAtomicAdd(p + 3, acc.w);
      acc = make_float4(0.f, 0.f, 0.f, 0.f);
      cur = d;
    }
    const float4 v = *(const float4*)(h1 + (long)src[e] * HID_C + c);
    acc.x += v.x;
    acc.y += v.y;
    acc.z += v.z;
    acc.w += v.w;
  }
  float* p = m + (long)cur * HID_C + c;
  unsafeAtomicAdd(p + 0, acc.x);
  unsafeAtomicAdd(p + 1, acc.y);
  unsafeAtomicAdd(p + 2, acc.z);
  unsafeAtomicAdd(p + 3, acc.w);
}

// h2 = relu(m * rsqrt(in_deg) + b_conv), in place on m.
__global__ void norm_bias_relu_kernel(float* __restrict__ m,
                                      const float* __restrict__ rin,
                                      const float* __restrict__ b_conv,
                                      long total) {
  const long i = (long)blockIdx.x * blockDim.x + threadIdx.x;
  if (i >= total) return;
  const int c = (int)(i & (HID_C - 1));
  const long node = i >> 7;  // HID_C == 128
  const float v = m[i] * rin[node] + b_conv[c];
  m[i] = fmaxf(v, 0.0f);
}

// ------------------------------------------------------------------ launch
extern "C" void kernel_launch(void* const* d_in, const int* in_sizes, int n_in,
                              void* d_out, int out_size, void* d_ws,
                              size_t ws_size, hipStream_t stream) {
  const float* x      = (const float*)d_in[0];
  const idx_t* src    = (const idx_t*)d_in[1];
  const idx_t* dst    = (const idx_t*)d_in[2];
  const float* W_conv = (const float*)d_in[3];  // [IN_C, HID_C] row-major
  const float* b_conv = (const float*)d_in[4];
  const float* W1     = (const float*)d_in[5];  // [HID_C, HID_C] (out,in)
  const float* b1     = (const float*)d_in[6];
  const float* W2     = (const float*)d_in[7];  // [OUT_C, HID_C] (out,in)
  const float* b2     = (const float*)d_in[8];
  float* out = (float*)d_out;

  const int N = in_sizes[0] / IN_C;  // 100000 (divisible by 16)
  const int E = in_sizes[1];         // 1600000

  // Workspace (floats):
  // deg_out[N] | deg_in[N] | Wc_t[HID_C*IN_C] | bufB[N*128] | bufA[N*128]
  float* ws      = (float*)d_ws;
  float* deg_out = ws;
  float* deg_in  = ws + N;
  float* Wc_t    = ws + 2L * N;                       // [HID_C, IN_C]
  float* bufB    = Wc_t + (long)HID_C * IN_C;         // m, then h2 (in place)
  float* bufA    = bufB + (long)N * HID_C;            // h1, then h3 (reused)

  // Zero deg_out + deg_in, and m, every call (graph-replay safe).
  zero_f32_kernel<<<64, 256, 0, stream>>>(ws, 2L * N);
  zero_f32_kernel<<<2048, 256, 0, stream>>>(bufB, (long)N * HID_C);

  transpose_kernel<<<(IN_C * HID_C + 255) / 256, 256, 0, stream>>>(
      W_conv, Wc_t, IN_C, HID_C);

  degree_kernel<<<(E + 255) / 256, 256, 0, stream>>>(src, dst, deg_out, deg_in, E);
  rsqrt_deg_kernel<<<(2 * N + 255) / 256, 256, 0, stream>>>(deg_out, 2 * N);

  const int rowTiles = N / 16;  // 6250 -> 3125 blocks of 2 waves

  // h1 = (x * rsqrt(out_deg)) @ W_conv        [N,256]@[256,128]
  gemm_wmma_kernel<IN_C, 8, HID_C, /*BIAS=*/false, /*RELU=*/false,
                   /*SCALE_A=*/true>
      <<<dim3(rowTiles / 2, 1), 64, 0, stream>>>(x, Wc_t, nullptr, deg_out,
                                                 bufA);

  // m[dst] += h1[src] over sorted edges
  const int waves = (E + EDGES_PER_WAVE - 1) / EDGES_PER_WAVE;
  aggregate_kernel<<<(waves * 32 + 255) / 256, 256, 0, stream>>>(bufA, src, dst,
                                                                 bufB, E);

  // h2 = relu(m * rsqrt(in_deg) + b_conv)
  norm_bias_relu_kernel<<<(int)(((long)N * HID_C + 255) / 256), 256, 0,
                          stream>>>(bufB, deg_in, b_conv, (long)N * HID_C);

  // h3 = relu(h2 @ W1^T + b1)                 [N,128]@[128,128]
  gemm_wmma_kernel<HID_C, 8, HID_C, /*BIAS=*/true, /*RELU=*/true,
                   /*SCALE_A=*/false>
      <<<dim3(rowTiles / 2, 1), 64, 0, stream>>>(bufB, W1, b1, nullptr, bufA);

  // out = h3 @ W2^T + b2                      [N,128]@[128,256]
  gemm_wmma_kernel<HID_C, 8, OUT_C, /*BIAS=*/true, /*RELU=*/false,
                   /*SCALE_A=*/false>
      <<<dim3(rowTiles / 2, 2), 64, 0, stream>>>(bufA, W2, b2, nullptr, out);
}